// LearnedDMemBP_74895639707838
// MI455X (gfx1250) — compile-verified
//
#include <hip/hip_runtime.h>
#include <stdint.h>

// Problem constants (from reference)
#define BB    32
#define MM    4096
#define NN    8192
#define DCQ   10
#define NITER 8
#define ALPHA     100.0f
#define INV_TEMP  100.0f      // 1/TEMP, TEMP = 0.01
#define BIGV      100000000.0f

#define CHK_TILE 256          // checks per block

typedef unsigned int u32x4 __attribute__((ext_vector_type(4)));
typedef int          i32x8 __attribute__((ext_vector_type(8)));
typedef int          i32x4 __attribute__((ext_vector_type(4)));

// ---------------------------------------------------------------------------
// TDM: async-load a dense 2D tile (rows x DCQ dwords, row stride DCQ dwords)
// from global memory into LDS. Issued by one wave; tracked by TENSORcnt.
// D# layout per CDNA5 ISA ch.8 (group0 128b, group1 256b; groups 2/3 unused
// for 2D tensors). clang-23 builtin form: 6 args (g0, g1, g2, g3, g4, cpol).
// ---------------------------------------------------------------------------
__device__ __forceinline__ void tdm_load_tile_2d(uint32_t lds_byte_off,
                                                 const void* gptr, int rows) {
  uint64_t ga = (uint64_t)(uintptr_t)gptr;
  u32x4 g0;
  g0.x = 1u;                                            // count=1, user desc
  g0.y = lds_byte_off;                                  // lds_addr (bytes)
  g0.z = (uint32_t)ga;                                  // global_addr[31:0]
  g0.w = ((uint32_t)(ga >> 32) & 0x01FFFFFFu) | (2u << 30); // addr[56:32]|type=2

  i32x8 g1;
  g1.s0 = 0x00020000;                   // data_size=2 (4B), no multicast/pad
  g1.s1 = (int)(DCQ << 16);             // tensor_dim0[15:0] = DCQ
  g1.s2 = (int)(rows << 16);            // tensor_dim0 hi=0 | tensor_dim1 lo
  g1.s3 = (int)(DCQ << 16);             // tensor_dim1 hi=0 | tile_dim0 = DCQ
  g1.s4 = rows;                         // tile_dim1 = rows (tile_dim2 = 0)
  g1.s5 = DCQ;                          // tensor_dim0_stride = DCQ elements
  g1.s6 = 0;
  g1.s7 = 0;

  i32x4 z4 = {0, 0, 0, 0};              // groups 2/3 unused (2D tensor)
  i32x8 z8 = {0, 0, 0, 0, 0, 0, 0, 0};
  __builtin_amdgcn_tensor_load_to_lds(g0, g1, z4, z4, z8, 0);
}

// ---------------------------------------------------------------------------
// llrs base term:  iter 0 : prior[n]
//                  iter>0 : (1-gamma[n])*prior[n] + gamma[n]*llrs_prev[b,n]
// check kernel then atomically accumulates the scatter on top.
// ---------------------------------------------------------------------------
__global__ __launch_bounds__(256) void bp_init_kernel(
    float* __restrict__ dst, const float* __restrict__ prior,
    const float* __restrict__ gamma, const float* __restrict__ prev,
    int use_mem) {
  int idx = blockIdx.x * blockDim.x + threadIdx.x;
  if (idx >= BB * NN) return;
  int n = idx & (NN - 1);
  float p = prior[n];
  if (use_mem) {
    float g = gamma[n];
    dst[idx] = (1.0f - g) * p + g * prev[idx];
  } else {
    dst[idx] = p;
  }
}

// ---------------------------------------------------------------------------
// One thread per check node (b, m). TDM stages the block's chk_nbrs and c2v
// tiles into LDS; each thread does the dc=10 leave-one-out update and
// scatter-adds c2v into llrs_accum with global f32 atomics.
// ---------------------------------------------------------------------------
__global__ __launch_bounds__(CHK_TILE) void bp_check_kernel(
    const float* __restrict__ syn,        // (B, M)
    const float* __restrict__ prior,      // (N)
    const int*   __restrict__ nbrs,       // (M, DC)
    const float* __restrict__ llrs_prev,  // (B, N)  (unused on first iter)
    float*       __restrict__ c2v,        // (B, M, DC)  read old / write new
    float*       __restrict__ llrs_accum, // (B, N)  pre-filled with base term
    int first_iter) {
  __shared__ int   s_nbrs[CHK_TILE * DCQ];
  __shared__ float s_c2v [CHK_TILE * DCQ];

  const int m0 = blockIdx.x * CHK_TILE;
  const int b  = blockIdx.y;

  // --- stage tiles via Tensor Data Mover (wave 0 only) ---
  if (threadIdx.x < 32) {
    tdm_load_tile_2d((uint32_t)(uintptr_t)(void*)s_nbrs,
                     nbrs + (size_t)m0 * DCQ, CHK_TILE);
    if (!first_iter) {
      tdm_load_tile_2d((uint32_t)(uintptr_t)(void*)s_c2v,
                       c2v + ((size_t)b * MM + m0) * DCQ, CHK_TILE);
    }
    __builtin_amdgcn_s_wait_tensorcnt(0);
  }
  __syncthreads();

  const int t = threadIdx.x;
  const int m = m0 + t;

  int   idx[DCQ];
  float v[DCQ];
#pragma unroll
  for (int k = 0; k < DCQ; ++k) idx[k] = s_nbrs[t * DCQ + k];

  if (first_iter) {
#pragma unroll
    for (int k = 0; k < DCQ; ++k) v[k] = prior[idx[k]];
  } else {
    const float* lp = llrs_prev + (size_t)b * NN;
#pragma unroll
    for (int k = 0; k < DCQ; ++k) v[k] = lp[idx[k]] - s_c2v[t * DCQ + k];
  }

  // smooth sign + magnitude
  float s[DCQ], a[DCQ];
#pragma unroll
  for (int k = 0; k < DCQ; ++k) {
    float x = v[k];
    // tanh(ALPHA*x) = 1 - 2/(exp(2*ALPHA*x)+1)  (saturates cleanly in fp32)
    s[k] = 1.0f - 2.0f / (__expf(2.0f * ALPHA * x) + 1.0f);
    a[k] = fabsf(x);
  }

  // leave-one-out sign product via prefix/suffix products
  float pre[DCQ + 1];
  pre[0] = 1.0f;
#pragma unroll
  for (int k = 0; k < DCQ; ++k) pre[k + 1] = pre[k] * s[k];
  float prod_excl[DCQ];
  float suf = 1.0f;
#pragma unroll
  for (int j = DCQ - 1; j >= 0; --j) { prod_excl[j] = pre[j] * suf; suf *= s[j]; }

  // leave-one-out softmin: the per-j max-shift is gmin (j != argmin) or the
  // second-min (j == argmin) -> only two shifted exp-sums needed (20 exps,
  // not DC^2=100). Masked BIG entry contributes exp(-(BIG-min)/TEMP) == 0.
  float gmin = BIGV, smin = BIGV;
  int gidx = 0;
#pragma unroll
  for (int k = 0; k < DCQ; ++k) {
    if (a[k] < gmin) { smin = gmin; gmin = a[k]; gidx = k; }
    else if (a[k] < smin) { smin = a[k]; }
  }
  float e1[DCQ], S1 = 0.0f, T1 = 0.0f;
#pragma unroll
  for (int k = 0; k < DCQ; ++k) {
    float e = __expf((gmin - a[k]) * INV_TEMP);
    e1[k] = e; S1 += e; T1 += a[k] * e;
  }
  float S2 = 0.0f, T2 = 0.0f;
#pragma unroll
  for (int k = 0; k < DCQ; ++k) {
    if (k != gidx) {
      float e = __expf((smin - a[k]) * INV_TEMP);
      S2 += e; T2 += a[k] * e;
    }
  }

  const float sgn = 1.0f - 2.0f * syn[(size_t)b * MM + m];
  float* c2v_row = c2v + ((size_t)b * MM + m) * DCQ;
  float* acc     = llrs_accum + (size_t)b * NN;
#pragma unroll
  for (int j = 0; j < DCQ; ++j) {
    float num = (j == gidx) ? T2 : (T1 - a[j] * e1[j]);
    float den = (j == gidx) ? S2 : (S1 - e1[j]);
    float c = sgn * prod_excl[j] * (num / den);
    c2v_row[j] = c;
    atomicAdd(&acc[idx[j]], c);   // global_atomic_add_f32, L2-resident
  }
}

// ---------------------------------------------------------------------------
// Driver: 8 iterations of (init base term, check update + scatter).
// LLR buffers ping-pong between workspace and d_out so iteration 7 (the
// last) lands in d_out. ws usage: B*M*DC + B*N floats = 6 MB exactly.
// ---------------------------------------------------------------------------
extern "C" void kernel_launch(void* const* d_in, const int* in_sizes, int n_in,
                              void* d_out, int out_size, void* d_ws, size_t ws_size,
                              hipStream_t stream) {
  const float* syn   = (const float*)d_in[0];   // (B, M)
  const float* prior = (const float*)d_in[1];   // (N)
  const float* gamma = (const float*)d_in[2];   // (N)
  const int*   nbrs  = (const int*)d_in[3];     // (M, DC)

  float* c2v   = (float*)d_ws;                          // B*M*DC floats
  float* llrsA = c2v + (size_t)BB * MM * DCQ;           // B*N floats
  float* llrsB = (float*)d_out;                         // B*N floats

  const int init_blocks = (BB * NN + 255) / 256;
  const dim3 chk_grid(MM / CHK_TILE, BB);

  const float* prev = nullptr;
  for (int it = 0; it < NITER; ++it) {
    float* cur = (it & 1) ? llrsB : llrsA;   // iter 7 -> llrsB == d_out
    bp_init_kernel<<<init_blocks, 256, 0, stream>>>(
        cur, prior, gamma, prev, it > 0 ? 1 : 0);
    bp_check_kernel<<<chk_grid, CHK_TILE, 0, stream>>>(
        syn, prior, nbrs, prev, c2v, cur, it == 0 ? 1 : 0);
    prev = cur;
  }
}